// TaskAModel_89395449298956
// MI455X (gfx1250) — compile-verified
//
#include <hip/hip_runtime.h>

// ---------------------------------------------------------------------------
// Problem constants (match reference):  B=32, S=512, D=4096
//   out[b,t] = bias + sum_{s: seg[b,s]==t} dot(hs[b,s,:], W)
// Kernel 1: per-position dot products via V_WMMA_F32_16X16X4_F32 (exact f32).
// Kernel 2: deterministic segment-sum of scalars (sorted segments => spans).
// ---------------------------------------------------------------------------

typedef float v2f __attribute__((ext_vector_type(2)));
typedef float v8f __attribute__((ext_vector_type(8)));

#define B_DIM 32
#define S_DIM 512
#define D_DIM 4096
#define TILE_ROWS 16
#define KC 32                    // K-chunk staged in LDS per iteration
#define KSTRIDE (KC + 2)         // pad to keep 8B alignment + kill bank conflicts
#define WAVES_PER_BLOCK 4
#define THREADS (WAVES_PER_BLOCK * 32)
#define N_TILES ((B_DIM * S_DIM) / TILE_ROWS)       // 1024
#define N_BLOCKS (N_TILES / WAVES_PER_BLOCK)        // 256

// Kernel 1: d[row] = dot(hs[row, :], W)  for row in [0, B*S)
__global__ __launch_bounds__(THREADS)
void dot_wmma_kernel(const float* __restrict__ hs,
                     const float* __restrict__ W,
                     float* __restrict__ dvec)
{
    __shared__ float w_lds[D_DIM];                                   // 16 KB
    __shared__ float a_lds[WAVES_PER_BLOCK][TILE_ROWS][KSTRIDE];     // ~8.5 KB

    // Preload W once per block (coalesced float4).
    for (int i = threadIdx.x; i < D_DIM / 4; i += THREADS) {
        ((float4*)w_lds)[i] = ((const float4*)W)[i];
    }
    __syncthreads();

    const int wave = threadIdx.x >> 5;
    const int lane = threadIdx.x & 31;
    const int tile = blockIdx.x * WAVES_PER_BLOCK + wave;  // 0..1023
    const int row0 = tile * TILE_ROWS;                     // global row b*S+s

    const float* base = hs + (size_t)row0 * D_DIM;

    // A-matrix 16x4 f32 lane layout (ISA 7.12.2):
    //   lanes 0-15 : M=lane,    v0 -> K=0, v1 -> K=1
    //   lanes 16-31: M=lane-16, v0 -> K=2, v1 -> K=3
    const int m  = lane & 15;
    const int k0 = (lane >> 4) * 2;

    v8f c = {};  // 16x16 f32 accumulator (every column identical by construction)

    for (int kb = 0; kb < D_DIM; kb += KC) {
        // Stage 16 rows x KC floats into LDS. 128 float4 slots, 4 per lane;
        // 8 consecutive lanes cover 128 contiguous bytes of one row.
        #pragma unroll
        for (int i = 0; i < (TILE_ROWS * KC) / (32 * 4); ++i) {
            const int idx = i * 32 + lane;     // 0..127
            const int r   = idx >> 3;          // row 0..15  (KC/4 = 8 slots/row)
            const int c4  = idx & 7;           // float4 slot within row
            const float4 v = *(const float4*)(base + (size_t)r * D_DIM + kb + c4 * 4);
            float* p = &a_lds[wave][r][c4 * 4];
            p[0] = v.x; p[1] = v.y; p[2] = v.z; p[3] = v.w;
        }
        __syncthreads();  // cross-lane LDS visibility (buffers are wave-private)

        // 8 accumulating WMMAs consume the chunk: C += A(16x4) * B(4x16),
        // where B[k][n] = w[kb+kc+k] for all n (W replicated across columns).
        #pragma unroll
        for (int kc = 0; kc < KC; kc += 4) {
            v2f a, wv;
            a[0]  = a_lds[wave][m][kc + k0];
            a[1]  = a_lds[wave][m][kc + k0 + 1];
            wv[0] = w_lds[kb + kc + k0];        // same-address broadcast in each half
            wv[1] = w_lds[kb + kc + k0 + 1];
            c = __builtin_amdgcn_wmma_f32_16x16x4_f32(
                    /*neg_a=*/false, a, /*neg_b=*/false, wv,
                    /*c_mod=*/(short)0, c, /*reuse_a=*/false, /*reuse_b=*/false);
        }
        __syncthreads();
    }

    // D layout: lane L, VGPR v  ->  C[v + 8*(L>>4)][L&15]; all columns equal.
    // Lanes 0 and 16 (column N=0) write the 16 finished dot products.
    if ((lane & 15) == 0) {
        const int mbase = (lane >> 4) * 8;
        #pragma unroll
        for (int i = 0; i < 8; ++i) {
            dvec[row0 + mbase + i] = c[i];
        }
    }
}

// Kernel 2: out[b,t] = bias + sum of dvec over the contiguous span seg==t.
// Deterministic (ascending-s accumulation), writes every output element.
__global__ __launch_bounds__(S_DIM)
void segsum_kernel(const float* __restrict__ dvec,
                   const int* __restrict__ seg,
                   const float* __restrict__ bias,
                   float* __restrict__ out)
{
    __shared__ int   s_seg[S_DIM];
    __shared__ float s_d[S_DIM];

    const int b = blockIdx.x;
    const int t = threadIdx.x;

    s_seg[t] = seg[b * S_DIM + t];
    s_d[t]   = dvec[b * S_DIM + t];
    __syncthreads();

    // lower_bound of t in the sorted segment row
    int lo = 0, hi = S_DIM;
    while (lo < hi) {
        const int mid = (lo + hi) >> 1;
        if (s_seg[mid] < t) lo = mid + 1; else hi = mid;
    }

    float acc = bias[0];
    for (int s = lo; s < S_DIM && s_seg[s] == t; ++s) acc += s_d[s];

    out[b * S_DIM + t] = acc;
}

extern "C" void kernel_launch(void* const* d_in, const int* in_sizes, int n_in,
                              void* d_out, int out_size, void* d_ws, size_t ws_size,
                              hipStream_t stream)
{
    const float* hs   = (const float*)d_in[0];  // [32, 512, 4096] f32
    const int*   seg  = (const int*)  d_in[1];  // [32, 512] i32 (sorted per row)
    const float* W    = (const float*)d_in[2];  // [4096, 1] f32
    const float* bias = (const float*)d_in[3];  // [1] f32
    float*       out  = (float*)d_out;          // [32, 512, 1] f32
    float*       dvec = (float*)d_ws;           // 32*512 floats = 64 KB scratch

    dot_wmma_kernel<<<N_BLOCKS, THREADS, 0, stream>>>(hs, W, dvec);
    segsum_kernel<<<B_DIM, S_DIM, 0, stream>>>(dvec, seg, bias, out);
}